// SVDLinearCustomGrad_72395968741811
// MI455X (gfx1250) — compile-verified
//
#include <hip/hip_runtime.h>
#include <hip/hip_bf16.h>
#include <stdint.h>

// ---------------------------------------------------------------------------
// Rank-20 truncated-SVD linear layer for MI455X (gfx1250, wave32, WMMA).
//
// y = samples @ W20^T + bias,  W20 = rank-20 truncation of W (1024x1024).
// W20 = W * P, P = V20 V20^T = projector onto top-20 right singular subspace.
// So:  A = W*V20 (1024x20),  y = (S*V20)*A^T + bias.
//
// Pipeline (all fp32 math via v_wmma_f32_16x16x4_f32, fp64 only for G):
//   1) G = W^T W               (fp64 accumulate; 19/20 eigengap is tiny)
//   2) power iteration -> v1, lambda1 (dominant direction of U[0,1) matrix)
//   3) 80 x block subspace iteration (b=64) on deflated G' = G - l1 v1 v1^T
//      each iter: CholQR (1 WG) + V=Z*Minv (WMMA) + Z=G'V (WMMA)
//   4) Rayleigh-Ritz: T=V^T Z, 64x64 Jacobi eigensolver (1 WG), top-19
//   5) V20=[v1 | V*Q19]; A=W*V20; swizzle into WMMA B-fragment layout
//   6) fused streaming kernel: y = (S*V20)*A^T + bias  (bandwidth bound:
//      268MB @ 23.3TB/s ~ 11.5us; only ~4 GFLOP of WMMA work)
// ---------------------------------------------------------------------------

typedef float v2f __attribute__((ext_vector_type(2)));
typedef float v8f __attribute__((ext_vector_type(8)));

#define WMMA_F32_4(a, b, c) \
  __builtin_amdgcn_wmma_f32_16x16x4_f32(false, (a), false, (b), (short)0, (c), false, false)

constexpr int NDIM = 1024;   // in = out = 1024
constexpr int RPAD = 32;     // rank 20 padded to 32 (N must tile by 16)
constexpr int BB   = 64;     // subspace block size
constexpr int POW_ITERS = 7;
constexpr int SS_ITERS  = 80;
constexpr int JAC_SWEEPS = 12;

// -------------------------------- setup ------------------------------------

__global__ __launch_bounds__(256) void k_gram(const float* __restrict__ W,
                                              float* __restrict__ G) {
  int j = blockIdx.x * 16 + threadIdx.x;
  int i = blockIdx.y * 16 + threadIdx.y;
  double acc = 0.0;
  for (int k = 0; k < NDIM; ++k)
    acc += (double)W[k * NDIM + i] * (double)W[k * NDIM + j];
  G[i * NDIM + j] = (float)acc;
}

__device__ inline float hash_to_float(uint32_t x) {
  x *= 2654435761u; x ^= x >> 16; x *= 2246822519u; x ^= x >> 13;
  x *= 3266489917u; x ^= x >> 16;
  // high 24 bits -> uniform in [-0.5, 0.5)
  return ((float)(x >> 8) * (1.0f / 16777216.0f)) - 0.5f;
}

__global__ void k_init(float* __restrict__ Z, float* __restrict__ v1) {
  int i = blockIdx.x * blockDim.x + threadIdx.x;
  if (i < NDIM * BB) Z[i] = hash_to_float((uint32_t)i);
  if (i < NDIM) v1[i] = 0.03125f;  // all-ones start: big overlap with v1
}

__global__ __launch_bounds__(256) void k_gemv(const float* __restrict__ G,
                                              const float* __restrict__ v,
                                              float* __restrict__ z) {
  __shared__ float vs[NDIM];
  for (int t = threadIdx.x; t < NDIM; t += 256) vs[t] = v[t];
  __syncthreads();
  int wave = threadIdx.x >> 5, lane = threadIdx.x & 31;
  int row0 = blockIdx.x * 32 + wave * 4;
  for (int rr = 0; rr < 4; ++rr) {
    int r = row0 + rr;
    float p = 0.f;
    for (int c = lane; c < NDIM; c += 32) p += G[r * NDIM + c] * vs[c];
    for (int off = 16; off > 0; off >>= 1) p += __shfl_down(p, off, 32);
    if (lane == 0) z[r] = p;
  }
}

__global__ __launch_bounds__(1024) void k_norm(const float* __restrict__ z,
                                               float* __restrict__ v,
                                               float* __restrict__ lam) {
  __shared__ float red[32];
  __shared__ float nrm;
  int t = threadIdx.x;
  float x = z[t];
  float s = x * x;
  for (int off = 16; off > 0; off >>= 1) s += __shfl_down(s, off, 32);
  if ((t & 31) == 0) red[t >> 5] = s;
  __syncthreads();
  if (t < 32) {
    float w = red[t];
    for (int off = 16; off > 0; off >>= 1) w += __shfl_down(w, off, 32);
    if (t == 0) { nrm = sqrtf(w); lam[0] = nrm; }
  }
  __syncthreads();
  v[t] = x / nrm;
}

// ------------------------- CholQR (one workgroup) --------------------------
// In:  Z (1024x64), v1, lam.  Out: MF = (L^-1)^T in WMMA-B fragment layout,
// dv = lam * (v1^T Z) * M  (deflation coefficients for next G-apply).

__global__ __launch_bounds__(256) void k_cholqr(const float* __restrict__ Z,
                                                const float* __restrict__ v1,
                                                const float* __restrict__ lam,
                                                float* __restrict__ MF,
                                                float* __restrict__ dv) {
  __shared__ float Cs[64 * 65];
  __shared__ float Ms[64 * 65];
  __shared__ float tpart[8][64];
  __shared__ float ts[64];
  int tid = threadIdx.x, wave = tid >> 5, lane = tid & 31;
  int l16 = lane & 15, off = (lane >= 16) ? 2 : 0;

  // C = Z^T Z via WMMA: wave w -> M-tile mi=w>>1, N-tiles (w&1)*2+{0,1}
  {
    int mi = wave >> 1, ntb = (wave & 1) * 2;
    v8f a0 = {}, a1 = {};
    for (int k = 0; k < NDIM; k += 4) {
      const float* zr0 = Z + (size_t)(k + off) * BB;
      const float* zr1 = zr0 + BB;
      v2f av; av.x = zr0[mi * 16 + l16];         av.y = zr1[mi * 16 + l16];
      v2f b0; b0.x = zr0[ntb * 16 + l16];        b0.y = zr1[ntb * 16 + l16];
      v2f b1; b1.x = zr0[(ntb + 1) * 16 + l16];  b1.y = zr1[(ntb + 1) * 16 + l16];
      a0 = WMMA_F32_4(av, b0, a0);
      a1 = WMMA_F32_4(av, b1, a1);
    }
    int rbase = mi * 16 + ((lane >= 16) ? 8 : 0);
    for (int r = 0; r < 8; ++r) {
      Cs[(rbase + r) * 65 + ntb * 16 + l16]       = a0[r];
      Cs[(rbase + r) * 65 + (ntb + 1) * 16 + l16] = a1[r];
    }
  }

  // t = v1^T Z  (deterministic tree-sum: per-wave partials, fixed-order add)
  {
    float p0 = 0.f, p1 = 0.f;
    int rlo = wave * 128;
    for (int r = rlo; r < rlo + 128; ++r) {
      float vr = v1[r];
      p0 = fmaf(vr, Z[r * BB + lane], p0);
      p1 = fmaf(vr, Z[r * BB + lane + 32], p1);
    }
    tpart[wave][lane]      = p0;
    tpart[wave][lane + 32] = p1;
  }
  __syncthreads();
  if (tid < 64) {
    float s = 0.f;
    for (int w = 0; w < 8; ++w) s += tpart[w][tid];
    ts[tid] = s;
  }
  __syncthreads();

  // Cholesky C = L L^T (in place, lower)
  for (int k = 0; k < BB; ++k) {
    if (tid == 0) Cs[k * 65 + k] = sqrtf(fmaxf(Cs[k * 65 + k], 1e-30f));
    __syncthreads();
    if (tid > k && tid < BB) Cs[tid * 65 + k] /= Cs[k * 65 + k];
    __syncthreads();
    for (int e = tid; e < 4096; e += 256) {
      int i = e >> 6, j = e & 63;
      if (i > k && j > k && j <= i)
        Cs[i * 65 + j] -= Cs[i * 65 + k] * Cs[j * 65 + k];
    }
    __syncthreads();
  }

  // Ms[k][j] = Linv[j][k] (upper); forward substitution, i sequential
  for (int i = 0; i < BB; ++i) {
    if (tid < BB) {
      int j = tid;
      if (j <= i) {
        float s = (i == j) ? 1.0f : 0.0f;
        for (int k = j; k < i; ++k) s -= Cs[i * 65 + k] * Ms[j * 65 + k];
        Ms[j * 65 + i] = s / Cs[i * 65 + i];
      } else {
        Ms[j * 65 + i] = 0.0f;
      }
    }
    __syncthreads();
  }

  // d = lam * (t M);  MF = M in B-fragment swizzle
  float lamv = lam[0];
  if (tid < BB) {
    int j = tid;
    float s = 0.f;
    for (int k = 0; k <= j; ++k) s += ts[k] * Ms[k * 65 + j];
    dv[j] = lamv * s;
  }
  for (int e = tid; e < 4096; e += 256) {
    int v = e & 1, ln = (e >> 1) & 31, nt = (e >> 6) & 3, kq = e >> 8;
    MF[e] = Ms[(kq * 4 + v + ((ln >= 16) ? 2 : 0)) * 65 + nt * 16 + (ln & 15)];
  }
}

// ----------------------- V = Z * M (orthonormalize) ------------------------

__global__ __launch_bounds__(256) void k_applyM(const float* __restrict__ Z,
                                                const float* __restrict__ MF,
                                                float* __restrict__ V,
                                                float* __restrict__ VF) {
  __shared__ float plds[8][16][64];
  int tid = threadIdx.x, wave = tid >> 5, lane = tid & 31;
  int l16 = lane & 15, off = (lane >= 16) ? 2 : 0;
  int r0 = (blockIdx.x * 8 + wave) * 16;
  const v2f* MF2 = (const v2f*)MF;
  v8f acc[4] = {};
  for (int k = 0; k < BB; k += 4) {
    v2f av = *(const v2f*)(Z + (size_t)(r0 + l16) * BB + k + off);
    int base = (k >> 2) * 128 + lane;
    acc[0] = WMMA_F32_4(av, MF2[base],      acc[0]);
    acc[1] = WMMA_F32_4(av, MF2[base + 32], acc[1]);
    acc[2] = WMMA_F32_4(av, MF2[base + 64], acc[2]);
    acc[3] = WMMA_F32_4(av, MF2[base + 96], acc[3]);
  }
  int rb = (lane >= 16) ? 8 : 0;
  for (int nt = 0; nt < 4; ++nt)
    for (int r = 0; r < 8; ++r)
      plds[wave][rb + r][nt * 16 + l16] = acc[nt][r];
  __syncthreads();
  for (int s = 0; s < 32; ++s) {                 // plain V (row-major)
    int idx = s * 32 + lane;
    int row = idx >> 6, col = idx & 63;
    V[(size_t)(r0 + row) * BB + col] = plds[wave][row][col];
  }
  v2f* VF2 = (v2f*)VF;                           // fragment-swizzled V
  for (int kgl = 0; kgl < 4; ++kgl) {
    int krow = kgl * 4 + off;
    for (int nt = 0; nt < 4; ++nt) {
      v2f w;
      w.x = plds[wave][krow][nt * 16 + l16];
      w.y = plds[wave][krow + 1][nt * 16 + l16];
      VF2[(r0 / 4 + kgl) * 128 + nt * 32 + lane] = w;
    }
  }
}

// ----------------------- Z = G*V - v1*d^T (deflated) -----------------------

__global__ __launch_bounds__(256) void k_gemmG(const float* __restrict__ G,
                                               const float* __restrict__ VF,
                                               const float* __restrict__ v1,
                                               const float* __restrict__ dv,
                                               float* __restrict__ Z) {
  int tid = threadIdx.x, wave = tid >> 5, lane = tid & 31;
  int l16 = lane & 15, off = (lane >= 16) ? 2 : 0;
  int r0 = (blockIdx.x * 8 + wave) * 16;
  const v2f* VF2 = (const v2f*)VF;
  v8f acc[4] = {};
  const float* grow = G + (size_t)(r0 + l16) * NDIM + off;
  for (int k = 0; k < NDIM; k += 4) {
    v2f av = *(const v2f*)(grow + k);
    int base = (k >> 2) * 128 + lane;
    acc[0] = WMMA_F32_4(av, VF2[base],      acc[0]);
    acc[1] = WMMA_F32_4(av, VF2[base + 32], acc[1]);
    acc[2] = WMMA_F32_4(av, VF2[base + 64], acc[2]);
    acc[3] = WMMA_F32_4(av, VF2[base + 96], acc[3]);
  }
  int rbase = r0 + ((lane >= 16) ? 8 : 0);
  float v1v[8];
  for (int r = 0; r < 8; ++r) v1v[r] = v1[rbase + r];
  for (int nt = 0; nt < 4; ++nt) {
    float dn = dv[nt * 16 + l16];
    for (int r = 0; r < 8; ++r)
      Z[(size_t)(rbase + r) * BB + nt * 16 + l16] = acc[nt][r] - v1v[r] * dn;
  }
}

// -------------- Rayleigh-Ritz: T = V^T Z, Jacobi, top-19 -------------------

__global__ __launch_bounds__(256) void k_eig(const float* __restrict__ V,
                                             const float* __restrict__ Z,
                                             float* __restrict__ Qsel) {
  __shared__ float Ts[64 * 65];
  __shared__ float Qs[64 * 65];
  __shared__ float cs_[32], sn_[32];
  __shared__ int pp_[32], qq_[32];
  __shared__ int sel[19];
  int tid = threadIdx.x, wave = tid >> 5, lane = tid & 31;
  int l16 = lane & 15, off = (lane >= 16) ? 2 : 0;

  {  // T = V^T Z via WMMA
    int mi = wave >> 1, ntb = (wave & 1) * 2;
    v8f a0 = {}, a1 = {};
    for (int k = 0; k < NDIM; k += 4) {
      const float* vr0 = V + (size_t)(k + off) * BB;
      const float* vr1 = vr0 + BB;
      const float* zr0 = Z + (size_t)(k + off) * BB;
      const float* zr1 = zr0 + BB;
      v2f av; av.x = vr0[mi * 16 + l16];         av.y = vr1[mi * 16 + l16];
      v2f b0; b0.x = zr0[ntb * 16 + l16];        b0.y = zr1[ntb * 16 + l16];
      v2f b1; b1.x = zr0[(ntb + 1) * 16 + l16];  b1.y = zr1[(ntb + 1) * 16 + l16];
      a0 = WMMA_F32_4(av, b0, a0);
      a1 = WMMA_F32_4(av, b1, a1);
    }
    int rbase = mi * 16 + ((lane >= 16) ? 8 : 0);
    for (int r = 0; r < 8; ++r) {
      Ts[(rbase + r) * 65 + ntb * 16 + l16]       = a0[r];
      Ts[(rbase + r) * 65 + (ntb + 1) * 16 + l16] = a1[r];
    }
  }
  __syncthreads();
  for (int e = tid; e < 4096; e += 256) {  // symmetrize + Q = I
    int i = e >> 6, j = e & 63;
    if (i < j) {
      float a = 0.5f * (Ts[i * 65 + j] + Ts[j * 65 + i]);
      Ts[i * 65 + j] = a; Ts[j * 65 + i] = a;
    }
    Qs[i * 65 + j] = (i == j) ? 1.0f : 0.0f;
  }
  __syncthreads();

  for (int sweep = 0; sweep < JAC_SWEEPS; ++sweep) {
    for (int rnd = 0; rnd < 63; ++rnd) {  // tournament: 32 disjoint pairs
      if (tid < 32) {
        int p, q;
        if (tid == 0) { p = 63; q = rnd; }
        else { p = (rnd + tid) % 63; q = (rnd + 63 - tid) % 63; }
        if (p > q) { int t = p; p = q; q = t; }
        float app = Ts[p * 65 + p], aqq = Ts[q * 65 + q], apq = Ts[p * 65 + q];
        float c = 1.0f, s = 0.0f;
        if (fabsf(apq) > 1e-12f * (fabsf(app) + fabsf(aqq)) + 1e-36f) {
          float tau = (aqq - app) / (2.0f * apq);
          float t_ = ((tau >= 0.f) ? 1.0f : -1.0f) / (fabsf(tau) + sqrtf(1.0f + tau * tau));
          c = 1.0f / sqrtf(1.0f + t_ * t_);
          s = t_ * c;
        }
        cs_[tid] = c; sn_[tid] = s; pp_[tid] = p; qq_[tid] = q;
      }
      __syncthreads();
      for (int e = tid; e < 2048; e += 256) {  // rows: T <- J^T T
        int m = e >> 6, col = e & 63;
        int p = pp_[m], q = qq_[m];
        float c = cs_[m], s = sn_[m];
        float x = Ts[p * 65 + col], y = Ts[q * 65 + col];
        Ts[p * 65 + col] = c * x - s * y;
        Ts[q * 65 + col] = s * x + c * y;
      }
      __syncthreads();
      for (int e = tid; e < 2048; e += 256) {  // cols: T <- T J ; Q <- Q J
        int m = e >> 6, row = e & 63;
        int p = pp_[m], q = qq_[m];
        float c = cs_[m], s = sn_[m];
        float x = Ts[row * 65 + p], y = Ts[row * 65 + q];
        Ts[row * 65 + p] = c * x - s * y;
        Ts[row * 65 + q] = s * x + c * y;
        float qx = Qs[row * 65 + p], qy = Qs[row * 65 + q];
        Qs[row * 65 + p] = c * qx - s * qy;
        Qs[row * 65 + q] = s * qx + c * qy;
      }
      __syncthreads();
    }
  }

  if (tid == 0) {  // top-19 Ritz pairs
    unsigned long long used = 0ull;
    for (int r = 0; r < 19; ++r) {
      int best = 0; float bv = -1e30f;
      for (int j = 0; j < 64; ++j) {
        if (used & (1ull << j)) continue;
        float ev = Ts[j * 65 + j];
        if (ev > bv) { bv = ev; best = j; }
      }
      used |= (1ull << best);
      sel[r] = best;
    }
  }
  __syncthreads();
  for (int e = tid; e < 64 * 19; e += 256)
    Qsel[e] = Qs[(e / 19) * 65 + sel[e % 19]];
}

// ----------------- assemble V20, A = W*V20, swizzles -----------------------

__global__ __launch_bounds__(256) void k_v20(const float* __restrict__ v1,
                                             const float* __restrict__ V,
                                             const float* __restrict__ Qsel,
                                             float* __restrict__ V20) {
  int tid = threadIdx.x;
  int c = tid & 31;
  int row = blockIdx.x * 8 + (tid >> 5);
  float val = 0.f;
  if (c == 0) val = v1[row];
  else if (c < 20) {
    float s = 0.f;
    const float* vr = V + (size_t)row * BB;
    for (int k = 0; k < BB; ++k) s += vr[k] * Qsel[k * 19 + (c - 1)];
    val = s;
  }
  V20[(size_t)row * RPAD + c] = val;
}

__global__ __launch_bounds__(256) void k_amat(const float* __restrict__ W,
                                              const float* __restrict__ V20,
                                              float* __restrict__ A) {
  int tid = threadIdx.x;
  int j = tid & 31;
  int i = blockIdx.x * 8 + (tid >> 5);
  float s = 0.f;
  const float* wr = W + (size_t)i * NDIM;
  for (int k = 0; k < NDIM; ++k) s += wr[k] * V20[k * RPAD + j];
  A[(size_t)i * RPAD + j] = s;
}

__global__ void k_swz(const float* __restrict__ V20, const float* __restrict__ A,
                      float* __restrict__ V20F, float* __restrict__ AF) {
  int e = blockIdx.x * blockDim.x + threadIdx.x;
  if (e < 32768) {  // V20F: [kq 0..255][nt 0..1][lane][v]
    int v = e & 1, ln = (e >> 1) & 31, nt = (e >> 6) & 1, kq = e >> 7;
    V20F[e] = V20[(kq * 4 + v + ((ln >= 16) ? 2 : 0)) * RPAD + nt * 16 + (ln & 15)];
  }
  if (e < 20480) {  // AF: [ntile 0..63][kf 0..4][lane][v]  (B-frag of A^T)
    int v = e & 1, ln = (e >> 1) & 31;
    int rest = e >> 6, kf = rest % 5, ntile = rest / 5;
    AF[e] = A[(size_t)(ntile * 16 + (ln & 15)) * RPAD + kf * 4 + v + ((ln >= 16) ? 2 : 0)];
  }
}

// --------------- main streaming kernel: y = (S V20) A^T + b ----------------

__global__ __launch_bounds__(256) void k_main(const float* __restrict__ S,
                                              const float* __restrict__ V20F,
                                              const float* __restrict__ AF,
                                              const float* __restrict__ bias,
                                              float* __restrict__ Y) {
  __shared__ float plds[8][16][32];
  int tid = threadIdx.x, wave = tid >> 5, lane = tid & 31;
  int l16 = lane & 15, off = (lane >= 16) ? 2 : 0;
  int row0 = (blockIdx.x * 8 + wave) * 16;
  const v2f* VF2 = (const v2f*)V20F;
  const v2f* AF2 = (const v2f*)AF;

  // proj: P(16x32) = S_tile(16x1024) * V20pad(1024x32)
  v8f p0 = {}, p1 = {};
  const float* srow = S + (size_t)(row0 + l16) * NDIM + off;
  for (int k = 0; k < NDIM; k += 4) {
    v2f av = *(const v2f*)(srow + k);
    int base = (k >> 2) * 64 + lane;
    p0 = WMMA_F32_4(av, VF2[base],      p0);
    p1 = WMMA_F32_4(av, VF2[base + 32], p1);
  }
  int rb = (lane >= 16) ? 8 : 0;
  for (int r = 0; r < 8; ++r) {     // C/D-layout -> LDS -> A-fragments
    plds[wave][rb + r][l16]      = p0[r];
    plds[wave][rb + r][16 + l16] = p1[r];
  }
  __syncthreads();
  v2f pa[5];
  for (int kf = 0; kf < 5; ++kf)    // K = 20 = 5 k-frags exactly
    pa[kf] = *(const v2f*)&plds[wave][l16][kf * 4 + off];

  // expand: Y(16x1024) = P(16x20) * A^T(20x1024) + bias
  for (int nt = 0; nt < 64; ++nt) {
    int n0 = nt * 16;
    float bb = bias[n0 + l16];
    v8f acc = {bb, bb, bb, bb, bb, bb, bb, bb};
    const v2f* afn = AF2 + nt * 160;
    for (int kf = 0; kf < 5; ++kf)
      acc = WMMA_F32_4(pa[kf], afn[kf * 32 + lane], acc);
    int rowb = row0 + rb;
    for (int r = 0; r < 8; ++r)
      Y[(size_t)(rowb + r) * NDIM + n0 + l16] = acc[r];
  }
}

// ------------------------------- launcher ----------------------------------

extern "C" void kernel_launch(void* const* d_in, const int* in_sizes, int n_in,
                              void* d_out, int out_size, void* d_ws, size_t ws_size,
                              hipStream_t stream) {
  (void)in_sizes; (void)n_in; (void)out_size; (void)ws_size;
  const float* S    = (const float*)d_in[0];   // samples 32768x1024
  const float* W    = (const float*)d_in[1];   // weight 1024x1024
  const float* bias = (const float*)d_in[2];   // bias 1024
  float* Y  = (float*)d_out;
  float* ws = (float*)d_ws;

  float* G    = ws;               // 1048576
  float* Z    = G + 1048576;      // 65536
  float* V    = Z + 65536;        // 65536
  float* VF   = V + 65536;        // 65536
  float* MF   = VF + 65536;       // 4096
  float* dv   = MF + 4096;        // 64
  float* v1   = dv + 64;          // 1024
  float* lam  = v1 + 1024;        // 64 (padded)
  float* zvec = lam + 64;         // 1024
  float* Qsel = zvec + 1024;      // 1280 (padded)
  float* V20  = Qsel + 1280;      // 32768
  float* Apad = V20 + 32768;      // 32768
  float* V20F = Apad + 32768;     // 32768
  float* AF   = V20F + 32768;     // 20480  -> total ~5.5 MB

  k_gram<<<dim3(64, 64), dim3(16, 16), 0, stream>>>(W, G);
  k_init<<<256, 256, 0, stream>>>(Z, v1);
  for (int it = 0; it < POW_ITERS; ++it) {
    k_gemv<<<32, 256, 0, stream>>>(G, v1, zvec);
    k_norm<<<1, 1024, 0, stream>>>(zvec, v1, lam);
  }
  for (int it = 0; it < SS_ITERS; ++it) {
    k_cholqr<<<1, 256, 0, stream>>>(Z, v1, lam, MF, dv);
    k_applyM<<<8, 256, 0, stream>>>(Z, MF, V, VF);
    k_gemmG<<<8, 256, 0, stream>>>(G, VF, v1, dv, Z);
  }
  k_eig<<<1, 256, 0, stream>>>(V, Z, Qsel);
  k_v20<<<128, 256, 0, stream>>>(v1, V, Qsel, V20);
  k_amat<<<128, 256, 0, stream>>>(W, V20, Apad);
  k_swz<<<208, 256, 0, stream>>>(V20, Apad, V20F, AF);
  k_main<<<256, 256, 0, stream>>>(S, V20F, AF, bias, Y);
}